// GATModel_79568564126011
// MI455X (gfx1250) — compile-verified
//
#include <hip/hip_runtime.h>
#include <hip/hip_bf16.h>

#if defined(__HIP_DEVICE_COMPILE__) && !__has_builtin(__builtin_amdgcn_wmma_f32_16x16x4_f32)
#error "DEVICE: __builtin_amdgcn_wmma_f32_16x16x4_f32 not available on gfx1250"
#endif

typedef float v2f __attribute__((ext_vector_type(2)));
typedef float v8f __attribute__((ext_vector_type(8)));

// ---------------- problem constants ----------------
constexpr int NN   = 50000;          // nodes
constexpr int NE   = 200000;         // edges (before self loops)
constexpr int ETOT = NE + NN;        // edges + self loops
constexpr int FIN  = 74;             // node feature dim
constexpr int H1   = 4;              // heads layer 1
constexpr int D1   = 256;            // per-head dim
constexpr int HD1  = H1 * D1;        // 1024
constexpr int D2   = 256;            // layer-2 dim (H=1)
constexpr int NG   = 256;            // graphs
constexpr int ODIM = 12;
constexpr float NEG_SLOPE = 0.2f;

static inline int ceil_div_i(long long a, long long b) { return (int)((a + b - 1) / b); }

// ---------------- helpers ----------------
__device__ __forceinline__ void edge_sd(const int* __restrict__ ei, int idx, int& s, int& d) {
    if (idx < NE) { s = ei[idx]; d = ei[NE + idx]; }
    else          { int v = idx - NE; s = v; d = v; }   // self loop
}
// order-preserving float <-> uint map for atomic max
__device__ __forceinline__ unsigned fenc(float f) {
    unsigned u = __float_as_uint(f);
    return (u & 0x80000000u) ? ~u : (u | 0x80000000u);
}
__device__ __forceinline__ float fdec(unsigned u) {
    return __uint_as_float((u & 0x80000000u) ? (u ^ 0x80000000u) : ~u);
}
__device__ __forceinline__ float lrelu(float x) { return x > 0.f ? x : NEG_SLOPE * x; }

// ---------------- WMMA f32 GEMM: C[M,N] = op(A)[M,K] @ B[K,N] ----------------
// One wave computes a 16 x (16*NT) strip; A-fragment reused across NT B-tiles.
template<int NT, bool RELU_A>
__global__ __launch_bounds__(256)
void k_gemm_wmma(const float* __restrict__ A, const float* __restrict__ B,
                 float* __restrict__ C, int M, int N, int K)
{
    const int lane  = threadIdx.x & 31;
    const int wave  = threadIdx.x >> 5;
    const int wid   = blockIdx.x * 8 + wave;
    const int tilesN = N / (16 * NT);
    const int tilesM = M / 16;
    if (wid >= tilesM * tilesN) return;        // wave-uniform: EXEC stays full
    const int tm = wid / tilesN;
    const int tn = wid % tilesN;
    const int lm    = lane & 15;
    const int khalf = lane >> 4;               // 0: K rows {0,1}; 1: K rows {2,3}

    v8f acc[NT];
#pragma unroll
    for (int t = 0; t < NT; ++t) acc[t] = (v8f){0.f,0.f,0.f,0.f,0.f,0.f,0.f,0.f};

    const float* Arow = A + (size_t)(tm * 16 + lm) * K;
    const int colBase = tn * (16 * NT) + lm;
    const int kfull = K & ~3;

    int k0 = 0;
    for (; k0 < kfull; k0 += 4) {
        const int ka = k0 + 2 * khalf;
        float a0 = Arow[ka], a1 = Arow[ka + 1];
        if (RELU_A) { a0 = fmaxf(a0, 0.f); a1 = fmaxf(a1, 0.f); }
        v2f a; a.x = a0; a.y = a1;
#pragma unroll
        for (int t = 0; t < NT; ++t) {
            const int col = colBase + t * 16;
            v2f b;
            b.x = B[(size_t)ka * N + col];
            b.y = B[(size_t)(ka + 1) * N + col];
            acc[t] = __builtin_amdgcn_wmma_f32_16x16x4_f32(
                false, a, false, b, (short)0, acc[t], false, false);
        }
    }
    if (k0 < K) {                               // K % 4 tail, zero-padded
        const int ka = k0 + 2 * khalf;
        float a0 = (ka     < K) ? Arow[ka]     : 0.f;
        float a1 = (ka + 1 < K) ? Arow[ka + 1] : 0.f;
        if (RELU_A) { a0 = fmaxf(a0, 0.f); a1 = fmaxf(a1, 0.f); }
        v2f a; a.x = a0; a.y = a1;
#pragma unroll
        for (int t = 0; t < NT; ++t) {
            const int col = colBase + t * 16;
            v2f b;
            b.x = (ka     < K) ? B[(size_t)ka * N + col]       : 0.f;
            b.y = (ka + 1 < K) ? B[(size_t)(ka + 1) * N + col] : 0.f;
            acc[t] = __builtin_amdgcn_wmma_f32_16x16x4_f32(
                false, a, false, b, (short)0, acc[t], false, false);
        }
    }
    // C/D layout: VGPR r -> row r (lanes 0-15) / row 8+r (lanes 16-31)
    const int rowBase = tm * 16 + khalf * 8;
#pragma unroll
    for (int t = 0; t < NT; ++t) {
        const int col = colBase + t * 16;
#pragma unroll
        for (int r = 0; r < 8; ++r)
            C[(size_t)(rowBase + r) * N + col] = acc[t][r];
    }
}

// ---------------- init ----------------
__global__ void k_fill(float* __restrict__ p, float v, long long n) {
    long long i = (long long)blockIdx.x * blockDim.x + threadIdx.x;
    if (i < n) p[i] = v;
}
__global__ void k_bias_init(float* __restrict__ out, const float* __restrict__ b,
                            long long total, int F) {
    long long i = (long long)blockIdx.x * blockDim.x + threadIdx.x;
    if (i < total) out[i] = b[i & (F - 1)];     // F is a power of two (1024 / 256)
}

// ---------------- attention logits: als/ald[n,h] = sum_d h[n,h,d]*a[h,d] ----------------
__global__ void k_logits(const float* __restrict__ h, const float* __restrict__ a_src,
                         const float* __restrict__ a_dst,
                         float* __restrict__ als, float* __restrict__ ald, int H, int D)
{
    int i = blockIdx.x * blockDim.x + threadIdx.x;
    if (i >= NN * H) return;
    int n = i / H, hh = i - n * H;
    const float4* hv = (const float4*)(h + (size_t)n * H * D + (size_t)hh * D);
    const float4* sv = (const float4*)(a_src + (size_t)hh * D);
    const float4* dv = (const float4*)(a_dst + (size_t)hh * D);
    float ss = 0.f, sd = 0.f;
    for (int d = 0; d < D / 4; ++d) {
        float4 x = hv[d], s4 = sv[d], d4 = dv[d];
        ss += x.x * s4.x + x.y * s4.y + x.z * s4.z + x.w * s4.w;
        sd += x.x * d4.x + x.y * d4.y + x.z * d4.z + x.w * d4.w;
    }
    als[i] = ss; ald[i] = sd;
}

// ---------------- edge passes ----------------
__global__ void k_edge_max(const int* __restrict__ ei, const float* __restrict__ als,
                           const float* __restrict__ ald, unsigned* __restrict__ m, int H)
{
    int i = blockIdx.x * blockDim.x + threadIdx.x;
    if (i >= ETOT * H) return;
    int e = i / H, hh = i - e * H;
    int s, d; edge_sd(ei, e, s, d);
    float ev = lrelu(als[s * H + hh] + ald[d * H + hh]);
    atomicMax(&m[d * H + hh], fenc(ev));
}
__global__ void k_edge_den(const int* __restrict__ ei, const float* __restrict__ als,
                           const float* __restrict__ ald, const unsigned* __restrict__ m,
                           float* __restrict__ den, int H)
{
    int i = blockIdx.x * blockDim.x + threadIdx.x;
    if (i >= ETOT * H) return;
    int e = i / H, hh = i - e * H;
    int s, d; edge_sd(ei, e, s, d);
    float ev = lrelu(als[s * H + hh] + ald[d * H + hh]);
    atomicAdd(&den[d * H + hh], expf(ev - fdec(m[d * H + hh])));
}
// scatter: out[dst] += h[src]*alpha ; tpe threads per edge, 4 floats per thread
__global__ void k_edge_msg(const int* __restrict__ ei, const float* __restrict__ h,
                           const float* __restrict__ als, const float* __restrict__ ald,
                           const unsigned* __restrict__ m, const float* __restrict__ den,
                           float* __restrict__ out, int H, int D, int tpe)
{
    long long i = (long long)blockIdx.x * blockDim.x + threadIdx.x;
    int e = (int)(i / tpe);
    if (e >= ETOT) return;
    int c4 = (int)(i - (long long)e * tpe) * 4;     // column base within [0, H*D)
    int hh = c4 / D;
    int s, d; edge_sd(ei, e, s, d);
    float ev = lrelu(als[s * H + hh] + ald[d * H + hh]);
    float alpha = expf(ev - fdec(m[d * H + hh])) / (den[d * H + hh] + 1e-16f);
    const float4 hv = *(const float4*)(h + (size_t)s * H * D + c4);
    float* o = out + (size_t)d * H * D + c4;
    atomicAdd(o + 0, hv.x * alpha);
    atomicAdd(o + 1, hv.y * alpha);
    atomicAdd(o + 2, hv.z * alpha);
    atomicAdd(o + 3, hv.w * alpha);
}

// ---------------- pooling + FC ----------------
__global__ void k_pool(const float* __restrict__ out2, const int* __restrict__ batch,
                       float* __restrict__ pooled)
{
    long long i = (long long)blockIdx.x * blockDim.x + threadIdx.x;
    if (i >= (long long)NN * D2) return;
    int n = (int)(i / D2), j = (int)(i - (long long)n * D2);
    float v = fmaxf(out2[i], 0.f);
    atomicAdd(&pooled[(size_t)batch[n] * D2 + j], v);
}
__global__ void k_count(const int* __restrict__ batch, float* __restrict__ cnt)
{
    int n = blockIdx.x * blockDim.x + threadIdx.x;
    if (n < NN) atomicAdd(&cnt[batch[n]], 1.0f);
}
__global__ void k_fc(const float* __restrict__ pooled, const float* __restrict__ cnt,
                     const float* __restrict__ W, const float* __restrict__ bias,
                     float* __restrict__ out)
{
    int i = blockIdx.x * blockDim.x + threadIdx.x;
    if (i >= NG * ODIM) return;
    int g = i / ODIM, o = i - g * ODIM;
    float c = fmaxf(cnt[g], 1.0f);
    float s = 0.f;
    for (int k = 0; k < D2; ++k) s += pooled[(size_t)g * D2 + k] * W[k * ODIM + o];
    out[i] = s / c + bias[o];
}

// ---------------- launcher ----------------
extern "C" void kernel_launch(void* const* d_in, const int* in_sizes, int n_in,
                              void* d_out, int out_size, void* d_ws, size_t ws_size,
                              hipStream_t stream)
{
    const float* x    = (const float*)d_in[0];
    const int*   ei   = (const int*)  d_in[1];   // [2, NE]: src row then dst row
    const int*   bat  = (const int*)  d_in[2];
    const float* W1   = (const float*)d_in[3];
    const float* as1  = (const float*)d_in[4];
    const float* ad1  = (const float*)d_in[5];
    const float* b1   = (const float*)d_in[6];
    const float* W2   = (const float*)d_in[7];
    const float* as2  = (const float*)d_in[8];
    const float* ad2  = (const float*)d_in[9];
    const float* b2   = (const float*)d_in[10];
    const float* fcW  = (const float*)d_in[11];
    const float* fcb  = (const float*)d_in[12];
    float* out = (float*)d_out;

    // workspace layout (floats)
    float* ws = (float*)d_ws;
    size_t off = 0;
    float* regA = ws;             off += (size_t)NN * HD1;   // h1; later h2 + out2
    float* out1 = ws + off;       off += (size_t)NN * HD1;   // layer-1 aggregated output
    float* als1 = ws + off;       off += (size_t)NN * H1;
    float* ald1 = ws + off;       off += (size_t)NN * H1;
    unsigned* m1 = (unsigned*)(ws + off); off += (size_t)NN * H1;
    float* den1 = ws + off;       off += (size_t)NN * H1;
    float* als2 = ws + off;       off += NN;
    float* ald2 = ws + off;       off += NN;
    unsigned* m2 = (unsigned*)(ws + off); off += NN;
    float* den2 = ws + off;       off += NN;
    float* pooled = ws + off;     off += (size_t)NG * D2;
    float* cnt  = ws + off;       off += NG;
    float* h1   = regA;
    float* h2   = regA;                           // h1 dead once GEMM2 starts
    float* out2 = regA + (size_t)NN * D2;

    const int T = 256;
    constexpr int NT = 8;                          // N-tiles per wave (16*8 = 128 cols)

    // ---- layer 1 ----
    {   // h1 = x @ W1   (WMMA f32)
        int waves = (NN / 16) * (HD1 / (16 * NT));
        k_gemm_wmma<NT, false><<<ceil_div_i(waves, 8), T, 0, stream>>>(x, W1, h1, NN, HD1, FIN);
    }
    k_logits<<<ceil_div_i((long long)NN * H1, T), T, 0, stream>>>(h1, as1, ad1, als1, ald1, H1, D1);
    k_fill<<<ceil_div_i((long long)NN * H1, T), T, 0, stream>>>((float*)m1, 0.0f, (long long)NN * H1);
    k_fill<<<ceil_div_i((long long)NN * H1, T), T, 0, stream>>>(den1, 0.0f, (long long)NN * H1);
    k_edge_max<<<ceil_div_i((long long)ETOT * H1, T), T, 0, stream>>>(ei, als1, ald1, m1, H1);
    k_edge_den<<<ceil_div_i((long long)ETOT * H1, T), T, 0, stream>>>(ei, als1, ald1, m1, den1, H1);
    k_bias_init<<<ceil_div_i((long long)NN * HD1, T), T, 0, stream>>>(out1, b1, (long long)NN * HD1, HD1);
    k_edge_msg<<<ceil_div_i((long long)ETOT * (HD1 / 4), T), T, 0, stream>>>(
        ei, h1, als1, ald1, m1, den1, out1, H1, D1, HD1 / 4);

    // ---- layer 2 ----
    {   // h2 = relu(out1) @ W2   (WMMA f32, ReLU fused into A loads)
        int waves = (NN / 16) * (D2 / (16 * NT));
        k_gemm_wmma<NT, true><<<ceil_div_i(waves, 8), T, 0, stream>>>(out1, W2, h2, NN, D2, HD1);
    }
    k_logits<<<ceil_div_i(NN, T), T, 0, stream>>>(h2, as2, ad2, als2, ald2, 1, D2);
    k_fill<<<ceil_div_i(NN, T), T, 0, stream>>>((float*)m2, 0.0f, NN);
    k_fill<<<ceil_div_i(NN, T), T, 0, stream>>>(den2, 0.0f, NN);
    k_edge_max<<<ceil_div_i(ETOT, T), T, 0, stream>>>(ei, als2, ald2, m2, 1);
    k_edge_den<<<ceil_div_i(ETOT, T), T, 0, stream>>>(ei, als2, ald2, m2, den2, 1);
    k_bias_init<<<ceil_div_i((long long)NN * D2, T), T, 0, stream>>>(out2, b2, (long long)NN * D2, D2);
    k_edge_msg<<<ceil_div_i((long long)ETOT * (D2 / 4), T), T, 0, stream>>>(
        ei, h2, als2, ald2, m2, den2, out2, 1, D2, D2 / 4);

    // ---- relu + mean pool + FC ----
    k_fill<<<ceil_div_i((long long)NG * D2 + NG, T), T, 0, stream>>>(pooled, 0.0f, (long long)NG * D2 + NG);
    k_pool<<<ceil_div_i((long long)NN * D2, T), T, 0, stream>>>(out2, bat, pooled);
    k_count<<<ceil_div_i(NN, T), T, 0, stream>>>(bat, cnt);
    k_fc<<<ceil_div_i(NG * ODIM, T), T, 0, stream>>>(pooled, cnt, fcW, fcb, out);
    (void)in_sizes; (void)n_in; (void)out_size; (void)ws_size;
}